// DGCNNObjectEncoder_49409303773211
// MI455X (gfx1250) — compile-verified
//
#include <hip/hip_runtime.h>
#include <hip/hip_bf16.h>

typedef __attribute__((ext_vector_type(2))) float v2f;
typedef __attribute__((ext_vector_type(8))) float v8f;

#define NPTS 2048
#define NBATCH 16
#define KNN 20

// ---------------------------------------------------------------------------
// Squared norms per point: sq[b][m] = |x_m|^2  (tiny, bandwidth-trivial)
// ---------------------------------------------------------------------------
__global__ __launch_bounds__(256) void sqnorm_kernel(const float* __restrict__ x,
                                                     float* __restrict__ sq, int C) {
  const int b = blockIdx.y;
  const int m = blockIdx.x * 256 + threadIdx.x;
  const float* xm = x + ((size_t)b * NPTS + m) * C;
  float s = 0.0f;
  for (int c = 0; c < C; ++c) s = fmaf(xm[c], xm[c], s);
  sq[(size_t)b * NPTS + m] = s;
}

// ---------------------------------------------------------------------------
// kNN via WMMA gram matrix: one workgroup = 16 query rows.
// 8 waves sweep 128 column tiles: g = x_n . x_m via v_wmma_f32_16x16x4_f32,
// d = |x_n|^2 + |x_m|^2 - 2g written to a 16 x NPTS LDS strip (pitch padded
// +4 so the two half-waves hit disjoint bank halves). Then each wave runs the
// 20-round wave-wide arg-min on 2 rows (lane owns a strided 1/32 slice ->
// no cross-lane LDS hazards).
// A lane layout (16x4 f32): lane L: M=L&15, K = 2*(L>=16)+{0,1}.
// B lane layout (4x16):     lane L: N=L&15, K = 2*(L>=16)+{0,1}.
// C/D: VGPR r -> M = r + 8*(L>=16), N = L&15.
// ---------------------------------------------------------------------------
template <int C>
__global__ __launch_bounds__(256) void knn_wmma(const float* __restrict__ x,
                                                const float* __restrict__ sq,
                                                int* __restrict__ idx) {
  constexpr int CP = (C + 3) & ~3;  // pad K to multiple of 4 (C=3 -> 4)
  constexpr int KS = CP / 4;
  constexpr int NW = 8;
  constexpr int DPITCH = NPTS + 4;  // bank-conflict-free row pitch
  const int b = blockIdx.y;
  const int n0 = blockIdx.x * 16;
  const int tid = threadIdx.x, lane = tid & 31, wave = tid >> 5;
  const int col = lane & 15, hi = lane >> 4;

  __shared__ float dist[16 * DPITCH];  // ~128 KB (of 320 KB WGP LDS)

  const float* xb = x + (size_t)b * NPTS * C;
  const float* sqb = sq + (size_t)b * NPTS;
  // A operand base: query row n0+col, k-offset 2*hi
  const float* arow = xb + (size_t)(n0 + col) * C + 2 * hi;

  float sqn[8];
#pragma unroll
  for (int r = 0; r < 8; ++r) sqn[r] = sqb[n0 + r + 8 * hi];

  // guarded pair loader (guard only live for C % 4 != 0, i.e. C == 3)
  auto ldp = [&](const float* base, int ks) -> v2f {
    v2f r;
    const int k0 = ks * 4 + 2 * hi;
    r[0] = (k0 < C) ? base[ks * 4] : 0.0f;
    r[1] = (k0 + 1 < C) ? base[ks * 4 + 1] : 0.0f;
    return r;
  };

  for (int ct = wave; ct < NPTS / 16; ct += NW) {
    const int m0 = ct * 16;
    const float* brow = xb + (size_t)(m0 + col) * C + 2 * hi;
    v8f g0 = {}, g1 = {};
#pragma unroll 4
    for (int ks = 0; ks + 1 < KS; ks += 2) {  // two independent WMMA chains
      g0 = __builtin_amdgcn_wmma_f32_16x16x4_f32(false, ldp(arow, ks), false,
                                                 ldp(brow, ks), (short)0, g0,
                                                 false, false);
      g1 = __builtin_amdgcn_wmma_f32_16x16x4_f32(false, ldp(arow, ks + 1), false,
                                                 ldp(brow, ks + 1), (short)0, g1,
                                                 false, false);
    }
    if constexpr (KS & 1)
      g0 = __builtin_amdgcn_wmma_f32_16x16x4_f32(false, ldp(arow, KS - 1), false,
                                                 ldp(brow, KS - 1), (short)0, g0,
                                                 false, false);
    v8f g = g0 + g1;
    const float sqm = sqb[m0 + col];
#pragma unroll
    for (int r = 0; r < 8; ++r) {
      const int row = r + 8 * hi;
      dist[row * DPITCH + m0 + col] = sqn[r] + sqm - 2.0f * g[r];
    }
  }
  __syncthreads();

  // selection: wave handles rows 2*wave and 2*wave+1
  for (int rr = 0; rr < 2; ++rr) {
    const int row = wave * 2 + rr;
    float* drow = dist + row * DPITCH;
    int* irow = idx + ((size_t)b * NPTS + n0 + row) * KNN;
    for (int kk = 0; kk < KNN; ++kk) {
      float best = __builtin_inff();
      int bi = NPTS;
      for (int t2 = 0; t2 < NPTS / 32; ++t2) {
        const int m = lane + t2 * 32;
        const float d = drow[m];
        if (d < best || (d == best && m < bi)) { best = d; bi = m; }
      }
#pragma unroll
      for (int off = 16; off > 0; off >>= 1) {
        const float ov = __shfl_xor(best, off, 32);
        const int oi = __shfl_xor(bi, off, 32);
        if (ov < best || (ov == best && oi < bi)) { best = ov; bi = oi; }
      }
      if (lane == 0) irow[kk] = bi;
      if ((bi & 31) == lane) drow[bi] = __builtin_inff();  // owner lane kills it
    }
  }
}

// ---------------------------------------------------------------------------
// Fused EdgeConv: gather [nb-ctr; ctr] -> f32 WMMA GEMM -> BN -> LeakyReLU ->
// running max over k. One workgroup = 16 points. Two independent WMMA
// accumulator chains: TPW==2 interleaves the two output tiles (shares the
// B operand), TPW==1 splits K even/odd.
// ---------------------------------------------------------------------------
template <int CIN, int COUT, int NWAVES>
__global__ __launch_bounds__(NWAVES * 32) void edge_conv_wmma(
    const float* __restrict__ x, const int* __restrict__ idx,
    const float* __restrict__ Wt, const float* __restrict__ gg,
    const float* __restrict__ bb, const float* __restrict__ mm,
    const float* __restrict__ vv, float* __restrict__ out) {
  constexpr int K2 = 2 * CIN;
  constexpr int K2P = (K2 + 3) & ~3;  // block 1: 6 -> 8
  constexpr int KSn = K2P / 4;
  constexpr int NT = 16;
  constexpr int TILES = COUT / 16;
  constexpr int TPW = TILES / NWAVES;  // 1 or 2
  constexpr int NTHR = NWAVES * 32;
  static_assert(TILES % NWAVES == 0 && TPW <= 2, "uniform tile split");

  const int b = blockIdx.y;
  const int n0 = blockIdx.x * NT;
  const int tid = threadIdx.x;
  const int lane = tid & 31;
  const int wave = tid >> 5;
  const int col = lane & 15;
  const int hi = lane >> 4;

  __shared__ float ctr[NT * CIN];
  __shared__ float ftile[K2P * NT];  // B tile, row K x col N
  __shared__ float scale_s[COUT];
  __shared__ float shift_s[COUT];

  for (int o = tid; o < COUT; o += NTHR) {
    const float sc = gg[o] * rsqrtf(vv[o] + 1e-5f);
    scale_s[o] = sc;
    shift_s[o] = bb[o] - mm[o] * sc;
  }
  for (int t = tid; t < NT * CIN; t += NTHR) {
    const int j = t / CIN, c = t % CIN;
    ctr[j * CIN + c] = x[((size_t)b * NPTS + n0 + j) * CIN + c];
  }
  if (K2P != K2) {
    for (int t = tid; t < (K2P - K2) * NT; t += NTHR)
      ftile[(K2 + t / NT) * NT + (t % NT)] = 0.0f;  // zero pad rows (never rewritten)
  }
  __syncthreads();

  auto lda = [&](const float* wrow, int ks) -> v2f {  // guarded A pair
    v2f r;
    const int k0 = ks * 4 + 2 * hi;
    r[0] = (k0 < K2) ? wrow[k0] : 0.0f;
    r[1] = (k0 + 1 < K2) ? wrow[k0 + 1] : 0.0f;
    return r;
  };
  auto ldb = [&](int ks) -> v2f {  // B pair from LDS (pad rows are zero)
    v2f r;
    const int k0 = ks * 4 + 2 * hi;
    r[0] = ftile[k0 * NT + col];
    r[1] = ftile[(k0 + 1) * NT + col];
    return r;
  };

  v8f maxacc[TPW];
#pragma unroll
  for (int t = 0; t < TPW; ++t)
#pragma unroll
    for (int r = 0; r < 8; ++r) maxacc[t][r] = -__builtin_inff();

  auto bnmax = [&](const v8f& acc, int ot, v8f& mx) {
#pragma unroll
    for (int r = 0; r < 8; ++r) {
      const int o = ot * 16 + 8 * hi + r;
      float y = acc[r] * scale_s[o] + shift_s[o];
      y = (y >= 0.0f) ? y : 0.2f * y;
      mx[r] = fmaxf(mx[r], y);
    }
  };

  for (int kk = 0; kk < KNN; ++kk) {
    __syncthreads();  // previous WMMA reads of ftile complete
    for (int t = tid; t < NT * CIN; t += NTHR) {
      const int j = t / CIN, c = t % CIN;
      const int nb = idx[((size_t)b * NPTS + n0 + j) * KNN + kk];
      const float nv = x[((size_t)b * NPTS + nb) * CIN + c];  // L2-resident gather
      const float cv = ctr[j * CIN + c];
      ftile[c * NT + j] = nv - cv;
      ftile[(CIN + c) * NT + j] = cv;
      if (kk + 1 < KNN && (c & 31) == 0) {  // prefetch next-k neighbor row
        const int nb2 = idx[((size_t)b * NPTS + n0 + j) * KNN + kk + 1];
        __builtin_prefetch(x + ((size_t)b * NPTS + nb2) * CIN + c, 0, 1);
      }
    }
    __syncthreads();

    if constexpr (TPW == 1) {
      const float* wrow = Wt + (size_t)(wave * 16 + col) * K2;
      v8f a0 = {}, a1 = {};
#pragma unroll 4
      for (int ks = 0; ks + 1 < KSn; ks += 2) {
        a0 = __builtin_amdgcn_wmma_f32_16x16x4_f32(false, lda(wrow, ks), false,
                                                   ldb(ks), (short)0, a0, false, false);
        a1 = __builtin_amdgcn_wmma_f32_16x16x4_f32(false, lda(wrow, ks + 1), false,
                                                   ldb(ks + 1), (short)0, a1, false, false);
      }
      if constexpr (KSn & 1)
        a0 = __builtin_amdgcn_wmma_f32_16x16x4_f32(false, lda(wrow, KSn - 1), false,
                                                   ldb(KSn - 1), (short)0, a0, false, false);
      bnmax(a0 + a1, wave, maxacc[0]);
    } else {
      const float* wrow0 = Wt + (size_t)(wave * 16 + col) * K2;
      const float* wrow1 = Wt + (size_t)((wave + NWAVES) * 16 + col) * K2;
      v8f a0 = {}, a1 = {};
#pragma unroll 4
      for (int ks = 0; ks < KSn; ++ks) {
        const v2f bv = ldb(ks);  // shared B operand
        a0 = __builtin_amdgcn_wmma_f32_16x16x4_f32(false, lda(wrow0, ks), false,
                                                   bv, (short)0, a0, false, false);
        a1 = __builtin_amdgcn_wmma_f32_16x16x4_f32(false, lda(wrow1, ks), false,
                                                   bv, (short)0, a1, false, false);
      }
      bnmax(a0, wave, maxacc[0]);
      bnmax(a1, wave + NWAVES, maxacc[1]);
    }
  }

#pragma unroll
  for (int t = 0; t < TPW; ++t) {
    const int ot = wave + t * NWAVES;
    const size_t base = ((size_t)b * NPTS + n0 + col) * COUT + ot * 16 + 8 * hi;
#pragma unroll
    for (int r = 0; r < 8; ++r) out[base + r] = maxacc[t][r];
  }
}

// ---------------------------------------------------------------------------
// Final: y = W5(256x512) * concat(x1..x4) per 16-point tile via WMMA
// (two interleaved tile chains), BN + LeakyReLU, max over 16 points (shfl),
// per-tile partial -> reduce kernel (deterministic, no float atomics).
// ---------------------------------------------------------------------------
__global__ __launch_bounds__(256) void final_gemm_wmma(
    const float* __restrict__ x1, const float* __restrict__ x2,
    const float* __restrict__ x3, const float* __restrict__ x4,
    const float* __restrict__ W5, const float* __restrict__ gg,
    const float* __restrict__ bb, const float* __restrict__ mm,
    const float* __restrict__ vv, float* __restrict__ partial) {
  constexpr int K2 = 512, NT = 16, NTHR = 256;
  const int b = blockIdx.y;
  const int n0 = blockIdx.x * NT;
  const int tid = threadIdx.x, lane = tid & 31, wave = tid >> 5;
  const int col = lane & 15, hi = lane >> 4;

  __shared__ float ftile[K2 * NT];  // 32 KB
  __shared__ float scale_s[256], shift_s[256];

  for (int o = tid; o < 256; o += NTHR) {
    const float sc = gg[o] * rsqrtf(vv[o] + 1e-5f);
    scale_s[o] = sc;
    shift_s[o] = bb[o] - mm[o] * sc;
  }
  for (int t = tid; t < K2 * NT; t += NTHR) {
    const int j = t / K2, c = t % K2;  // c-fast: coalesced global reads
    const float* src;
    int cc, Cw;
    if (c < 64)       { src = x1; cc = c;       Cw = 64; }
    else if (c < 128) { src = x2; cc = c - 64;  Cw = 64; }
    else if (c < 256) { src = x3; cc = c - 128; Cw = 128; }
    else              { src = x4; cc = c - 256; Cw = 256; }
    ftile[c * NT + j] = src[((size_t)b * NPTS + n0 + j) * Cw + cc];
  }
  __syncthreads();

  const float* wrow0 = W5 + (size_t)(wave * 16 + col) * K2;
  const float* wrow1 = W5 + (size_t)((wave + 8) * 16 + col) * K2;
  v8f a0 = {}, a1 = {};
#pragma unroll 4
  for (int ks = 0; ks < K2 / 4; ++ks) {
    const int kb = ks * 4 + 2 * hi;
    const v2f av0 = {wrow0[kb], wrow0[kb + 1]};
    const v2f av1 = {wrow1[kb], wrow1[kb + 1]};
    const v2f bv = {ftile[kb * NT + col], ftile[(kb + 1) * NT + col]};
    a0 = __builtin_amdgcn_wmma_f32_16x16x4_f32(false, av0, false, bv, (short)0,
                                               a0, false, false);
    a1 = __builtin_amdgcn_wmma_f32_16x16x4_f32(false, av1, false, bv, (short)0,
                                               a1, false, false);
  }

  const v8f accs[2] = {a0, a1};
#pragma unroll
  for (int t = 0; t < 2; ++t) {
    const int ot = wave + t * 8;
#pragma unroll
    for (int r = 0; r < 8; ++r) {
      const int o = ot * 16 + 8 * hi + r;
      float y = accs[t][r] * scale_s[o] + shift_s[o];
      y = (y >= 0.0f) ? y : 0.2f * y;
#pragma unroll
      for (int off = 1; off < 16; off <<= 1)  // max over 16 points (stays in half-wave)
        y = fmaxf(y, __shfl_xor(y, off, 32));
      if (col == 0)
        partial[((size_t)b * (NPTS / NT) + blockIdx.x) * 256 + o] = y;
    }
  }
}

__global__ __launch_bounds__(256) void final_reduce_kernel(
    const float* __restrict__ partial, float* __restrict__ out) {
  const int b = blockIdx.x, o = threadIdx.x;
  float mx = -__builtin_inff();
  for (int t = 0; t < NPTS / 16; ++t)
    mx = fmaxf(mx, partial[((size_t)b * (NPTS / 16) + t) * 256 + o]);
  out[b * 256 + o] = mx;
}

// ---------------------------------------------------------------------------
extern "C" void kernel_launch(void* const* d_in, const int* in_sizes, int n_in,
                              void* d_out, int out_size, void* d_ws, size_t ws_size,
                              hipStream_t stream) {
  const float* pc = (const float*)d_in[0];
  const float* W1 = (const float*)d_in[1];
  const float *g1 = (const float*)d_in[2], *b1 = (const float*)d_in[3],
              *m1 = (const float*)d_in[4], *v1 = (const float*)d_in[5];
  const float* W2 = (const float*)d_in[6];
  const float *g2 = (const float*)d_in[7], *b2 = (const float*)d_in[8],
              *m2 = (const float*)d_in[9], *v2 = (const float*)d_in[10];
  const float* W3 = (const float*)d_in[11];
  const float *g3 = (const float*)d_in[12], *b3 = (const float*)d_in[13],
              *m3 = (const float*)d_in[14], *v3 = (const float*)d_in[15];
  const float* W4 = (const float*)d_in[16];
  const float *g4 = (const float*)d_in[17], *b4 = (const float*)d_in[18],
              *m4 = (const float*)d_in[19], *v4 = (const float*)d_in[20];
  const float* W5 = (const float*)d_in[21];
  const float *g5 = (const float*)d_in[22], *b5 = (const float*)d_in[23],
              *m5 = (const float*)d_in[24], *v5 = (const float*)d_in[25];

  // workspace: features stored point-major (B,N,C)
  float* ws = (float*)d_ws;
  float* x1 = ws;
  float* x2 = x1 + (size_t)NBATCH * NPTS * 64;
  float* x3 = x2 + (size_t)NBATCH * NPTS * 64;
  float* x4 = x3 + (size_t)NBATCH * NPTS * 128;
  float* partial = x4 + (size_t)NBATCH * NPTS * 256;
  int* idx = (int*)(partial + (size_t)NBATCH * (NPTS / 16) * 256);
  float* sqbuf = (float*)(idx + (size_t)NBATCH * NPTS * KNN);

  const dim3 sqGrid(NPTS / 256, NBATCH);
  const dim3 knnGrid(NPTS / 16, NBATCH);
  const dim3 convGrid(NPTS / 16, NBATCH);

  sqnorm_kernel<<<sqGrid, 256, 0, stream>>>(pc, sqbuf, 3);
  knn_wmma<3><<<knnGrid, 256, 0, stream>>>(pc, sqbuf, idx);
  edge_conv_wmma<3, 64, 4><<<convGrid, 128, 0, stream>>>(pc, idx, W1, g1, b1, m1, v1, x1);

  sqnorm_kernel<<<sqGrid, 256, 0, stream>>>(x1, sqbuf, 64);
  knn_wmma<64><<<knnGrid, 256, 0, stream>>>(x1, sqbuf, idx);
  edge_conv_wmma<64, 64, 4><<<convGrid, 128, 0, stream>>>(x1, idx, W2, g2, b2, m2, v2, x2);

  sqnorm_kernel<<<sqGrid, 256, 0, stream>>>(x2, sqbuf, 64);
  knn_wmma<64><<<knnGrid, 256, 0, stream>>>(x2, sqbuf, idx);
  edge_conv_wmma<64, 128, 8><<<convGrid, 256, 0, stream>>>(x2, idx, W3, g3, b3, m3, v3, x3);

  sqnorm_kernel<<<sqGrid, 256, 0, stream>>>(x3, sqbuf, 64);
  knn_wmma<128><<<knnGrid, 256, 0, stream>>>(x3, sqbuf, idx);
  edge_conv_wmma<128, 256, 8><<<convGrid, 256, 0, stream>>>(x3, idx, W4, g4, b4, m4, v4, x4);

  final_gemm_wmma<<<convGrid, 256, 0, stream>>>(x1, x2, x3, x4, W5, g5, b5, m5, v5, partial);
  final_reduce_kernel<<<NBATCH, 256, 0, stream>>>(partial, (float*)d_out);
}